// lovasz_softmax_loss_63359357551237
// MI455X (gfx1250) — compile-verified
//
#include <hip/hip_runtime.h>
#include <hip/hip_bf16.h>
#include <stdint.h>

// Lovasz-Softmax loss, sort-free histogram-rank formulation for MI455X (gfx1250).
//   Pass 0: zero workspace (hist + per-class loss accumulators)
//   Pass A: softmax -> per-(pixel,class) error -> packed (count,fg) histogram (u64 atomics)
//   Pass B: per-class descending exclusive scan of 65536 bins, one wave32 per class,
//           intra-256 scan via chained V_WMMA_F32_16X16X4_F32 (matmul-scan identity
//           Y = T*X + (J*X)*U). Zeroes hist for reuse as arrival counters.
//   Pass C: recompute softmax, packed-u64 arrival atomic -> exact (rank, fg-rank),
//           closed-form Lovasz gradient contribution, LDS-reduced into loss[c].
//   Pass D: average over present classes -> d_out[0].

#define C_      19
#define B_      4
#define HW_     (512 * 512)      // 262144 = 2^18
#define P_      (B_ * HW_)       // 1048576
#define NBINS   65536
#define IGNORE_ 255

typedef float v2f __attribute__((ext_vector_type(2)));
typedef float v8f __attribute__((ext_vector_type(8)));

static __device__ __forceinline__ v8f wmma4(v2f a, v2f b, v8f c) {
  // V_WMMA_F32_16X16X4_F32: D = A(16x4) * B(4x16) + C(16x16), wave32.
  return __builtin_amdgcn_wmma_f32_16x16x4_f32(false, a, false, b, (short)0, c,
                                               false, false);
}

static __device__ __forceinline__ int err_bin(float e) {
  // e in [0,1]; float bits are monotone for non-negative floats.
  // bits(1.0f) = 0x3F800000 -> bin 65024 < NBINS.
  return (int)(__float_as_uint(e) >> 14);
}

static __device__ __forceinline__ void softmax19(const float* __restrict__ logits,
                                                 int p, float* prob) {
  const int b = p >> 18;           // HW_ = 2^18
  const int r = p & (HW_ - 1);
  float x[C_];
#pragma unroll
  for (int c = 0; c < C_; ++c)
    x[c] = logits[((size_t)(b * C_ + c) << 18) + r];
  float mx = x[0];
#pragma unroll
  for (int c = 1; c < C_; ++c) mx = fmaxf(mx, x[c]);
  float s = 0.f;
#pragma unroll
  for (int c = 0; c < C_; ++c) { x[c] = __expf(x[c] - mx); s += x[c]; }
  const float inv = 1.f / s;
#pragma unroll
  for (int c = 0; c < C_; ++c) prob[c] = x[c] * inv;
}

// ---------------- Pass 0: zero workspace ----------------
__global__ void lovasz_zero_ws(unsigned long long* __restrict__ hist,
                               float* __restrict__ loss) {
  const int i = blockIdx.x * blockDim.x + threadIdx.x;
  if (i < C_ * NBINS) hist[i] = 0ull;
  if (i < C_) loss[i] = 0.f;
}

// ---------------- Pass A: build per-class (count,fg) histogram ----------------
__global__ void lovasz_passA_hist(const float* __restrict__ logits,
                                  const int* __restrict__ gt,
                                  unsigned long long* __restrict__ hist) {
  const int p = blockIdx.x * blockDim.x + threadIdx.x;
  if (p >= P_) return;
  const int label = gt[p];
  if (label == IGNORE_) return;    // ignored pixels are inert (proof: suffix has e=0)
  float prob[C_];
  softmax19(logits, p, prob);
#pragma unroll
  for (int c = 0; c < C_; ++c) {
    const float fg = (label == c) ? 1.f : 0.f;
    const float e  = fabsf(fg - prob[c]);
    const unsigned long long v = (1ull << 32) | (unsigned long long)(label == c);
    atomicAdd(&hist[(size_t)c * NBINS + err_bin(e)], v);
  }
}

// ---------------- Pass B: per-class descending exclusive scan, WMMA-based ----------------
// One wave32 per class. 256 chunks of 256 bins, high bins first.
// Processing index m = 0..255 maps to bin = base + 255 - m, so a FORWARD
// exclusive prefix scan over m equals the descending exclusive suffix over bins.
// X[k][col] = s[col*16 + k] (16x16, column-major 16-segments).
// Inclusive scan: Y = T*X + (J*X)*U   (T: k<=i ones; J: ones; U: k<col ones).
__global__ void lovasz_passB_scan_wmma(unsigned long long* __restrict__ hist,
                                       float2* __restrict__ cum,
                                       float* __restrict__ gts) {
  const int c    = blockIdx.x;
  const int lane = threadIdx.x;      // 0..31, wave32, EXEC all ones (WMMA requirement)
  const int sub  = lane >> 4;        // 0: lanes 0-15, 1: lanes 16-31
  const int nIdx = lane & 15;        // row index (A) / column index (B,D)

  unsigned long long* h = hist + (size_t)c * NBINS;
  float2* cm = cum + (size_t)c * NBINS;

  __shared__ float s_cnt[256];
  __shared__ float s_fg[256];

  float carry_cnt = 0.f, carry_fg = 0.f;

  for (int chunk = 0; chunk < NBINS / 256; ++chunk) {
    const int base = NBINS - 256 * (chunk + 1);

    // Stage chunk into LDS in processing order m (reverse-bin order).
#pragma unroll
    for (int t = 0; t < 8; ++t) {
      const int m = lane + 32 * t;
      const unsigned long long v = h[base + 255 - m];
      s_cnt[m] = (float)(unsigned)(v >> 32);
      s_fg[m]  = (float)(unsigned)(v & 0xffffffffu);
    }
    __syncthreads();
    // Zero hist for reuse as pass-C arrival counters (values are safe in LDS).
#pragma unroll
    for (int t = 0; t < 8; ++t) h[base + 255 - (lane + 32 * t)] = 0ull;

    // ---- J*X : column totals replicated down rows ----
    v8f jxC = {}, jxF = {};
#pragma unroll
    for (int t = 0; t < 4; ++t) {
      const int k0 = 4 * t + 2 * sub;     // A lane slots: (k0, k0+1); B rows likewise
      v2f a; a.x = 1.f; a.y = 1.f;        // J slice: all ones
      v2f bC, bF;
      bC.x = s_cnt[nIdx * 16 + k0]; bC.y = s_cnt[nIdx * 16 + k0 + 1];
      bF.x = s_fg [nIdx * 16 + k0]; bF.y = s_fg [nIdx * 16 + k0 + 1];
      jxC = wmma4(a, bC, jxC);
      jxF = wmma4(a, bF, jxF);
    }

    // ---- Y = T*X ... ----
    v8f yC = {}, yF = {};
#pragma unroll
    for (int t = 0; t < 4; ++t) {
      const int k0 = 4 * t + 2 * sub;
      v2f a;                              // T slice: T[i][k] = (k <= i)
      a.x = (k0     <= nIdx) ? 1.f : 0.f;
      a.y = (k0 + 1 <= nIdx) ? 1.f : 0.f;
      v2f bC, bF;
      bC.x = s_cnt[nIdx * 16 + k0]; bC.y = s_cnt[nIdx * 16 + k0 + 1];
      bF.x = s_fg [nIdx * 16 + k0]; bF.y = s_fg [nIdx * 16 + k0 + 1];
      yC = wmma4(a, bC, yC);
      yF = wmma4(a, bF, yF);
    }

    // ---- ... + (J*X)*U ----
#pragma unroll
    for (int t = 0; t < 4; ++t) {
      const int k0 = 4 * t + 2 * sub;
      // A slice of (J*X): A[i][k] = colTotal[k]; colTotal[k] lives in jx D vgpr0 lane k.
      v2f aC, aF;
      aC.x = __shfl(jxC[0], k0);     aC.y = __shfl(jxC[0], k0 + 1);
      aF.x = __shfl(jxF[0], k0);     aF.y = __shfl(jxF[0], k0 + 1);
      v2f bU;                             // U slice: U[k][col] = (k < col)
      bU.x = (k0     < nIdx) ? 1.f : 0.f;
      bU.y = (k0 + 1 < nIdx) ? 1.f : 0.f;
      yC = wmma4(aC, bU, yC);
      yF = wmma4(aF, bU, yF);
    }

    // D layout: vgpr r -> M = r + 8*sub, N = nIdx -> m = nIdx*16 + r + 8*sub.
#pragma unroll
    for (int r = 0; r < 8; ++r) {
      const int m = nIdx * 16 + r + 8 * sub;
      const float exC = yC[r] - s_cnt[m];     // inclusive -> exclusive
      const float exF = yF[r] - s_fg[m];
      cm[base + 255 - m] = make_float2(carry_cnt + exC, carry_fg + exF);
    }
    // Chunk totals = inclusive value at m=255 -> Y[15][15] -> vgpr7 lane31.
    carry_cnt += __shfl(yC[7], 31);
    carry_fg  += __shfl(yF[7], 31);
    __syncthreads();
  }

  if (lane == 0) gts[c] = carry_fg;           // total foreground count for class c
}

// ---------------- Pass C: per-element Lovasz contributions ----------------
__global__ void lovasz_passC_loss(const float* __restrict__ logits,
                                  const int* __restrict__ gt,
                                  unsigned long long* __restrict__ arrival, // zeroed hist
                                  const float2* __restrict__ cum,
                                  const float* __restrict__ gts,
                                  float* __restrict__ loss) {
  __shared__ float part[C_];
  __shared__ float s_gts[C_];
  if (threadIdx.x < C_) {
    part[threadIdx.x]  = 0.f;
    s_gts[threadIdx.x] = gts[threadIdx.x];
  }
  __syncthreads();

  const int p = blockIdx.x * blockDim.x + threadIdx.x;
  if (p < P_) {
    const int label = gt[p];
    if (label != IGNORE_) {
      float prob[C_];
      softmax19(logits, p, prob);
#pragma unroll
      for (int c = 0; c < C_; ++c) {
        const float g = s_gts[c];
        if (g > 0.f) {                       // classes='present'
          const float fg = (label == c) ? 1.f : 0.f;
          const float e  = fabsf(fg - prob[c]);
          const int bin  = err_bin(e);
          const size_t idx = (size_t)c * NBINS + bin;
          // One packed atomic gives a consistent (within-bin rank, within-bin fg-rank).
          const unsigned long long old =
              atomicAdd(&arrival[idx], (1ull << 32) | (unsigned long long)(label == c));
          const float2 cx = cum[idx];
          const float n0 = cx.x + (float)(unsigned)(old >> 32);         // exclusive rank
          const float f0 = cx.y + (float)(unsigned)(old & 0xffffffffu); // exclusive fg
          // J(f,n) = 1 - (g - f) / (g + n - f);  n0==0 -> J0 = 0 automatically.
          const float J0 = 1.f - (g - f0) / (g + (n0 - f0));
          const float f1 = f0 + fg;
          const float n1 = n0 + 1.f;
          const float J1 = 1.f - (g - f1) / (g + (n1 - f1));
          atomicAdd(&part[c], e * (J1 - J0));
        }
      }
    }
  }
  __syncthreads();
  if (threadIdx.x < C_) atomicAdd(&loss[threadIdx.x], part[threadIdx.x]);
}

// ---------------- Pass D: average over present classes ----------------
__global__ void lovasz_finalize(const float* __restrict__ gts,
                                const float* __restrict__ loss,
                                float* __restrict__ out) {
  if (threadIdx.x == 0 && blockIdx.x == 0) {
    float s = 0.f, n = 0.f;
    for (int c = 0; c < C_; ++c)
      if (gts[c] > 0.f) { s += loss[c]; n += 1.f; }
    out[0] = s / fmaxf(n, 1.f);
  }
}

extern "C" void kernel_launch(void* const* d_in, const int* in_sizes, int n_in,
                              void* d_out, int out_size, void* d_ws, size_t ws_size,
                              hipStream_t stream) {
  const float* logits = (const float*)d_in[0];   // [4,19,512,512] f32
  const int*   gt     = (const int*)d_in[1];     // [4,512,512] int
  float*       out    = (float*)d_out;           // scalar

  // Workspace layout (~20 MB):
  //   hist  : C_*NBINS u64  (9.96 MB)  -- histogram, then arrival counters
  //   cum   : C_*NBINS f2   (9.96 MB)  -- descending exclusive (count, fg) scan
  //   gts   : C_ f32, loss : C_ f32
  char* ws = (char*)d_ws;
  const size_t HIST_BYTES = (size_t)C_ * NBINS * sizeof(unsigned long long);
  unsigned long long* hist = (unsigned long long*)ws;
  float2* cum  = (float2*)(ws + HIST_BYTES);
  float*  gts  = (float*)(ws + 2 * HIST_BYTES);
  float*  loss = gts + 64;

  const int nz = C_ * NBINS;
  lovasz_zero_ws<<<(nz + 255) / 256, 256, 0, stream>>>(hist, loss);
  lovasz_passA_hist<<<P_ / 256, 256, 0, stream>>>(logits, gt, hist);
  lovasz_passB_scan_wmma<<<C_, 32, 0, stream>>>(hist, cum, gts);
  lovasz_passC_loss<<<P_ / 256, 256, 0, stream>>>(logits, gt, hist, cum, gts, loss);
  lovasz_finalize<<<1, 32, 0, stream>>>(gts, loss, out);
}